// AveFilter_13786845020779
// MI455X (gfx1250) — compile-verified
//
#include <hip/hip_runtime.h>
#include <stdint.h>

// ---------------------------------------------------------------------------
// Depthwise 5x5 box blur, (16,8,512,512) f32, 'same' zero padding.
// Memory-bound: 268 MB total traffic @ 23.3 TB/s => ~11.5 us floor. Separable
// compute (~0.3 G VALU ops) is far below that, so the design centers on data
// movement:
//   * TDM tensor_load_to_lds streams a (64+4)x(64+4) halo tile into LDS
//     (iterate mode: one row/iteration, LDS pitch 68). Right padding
//     comes free from tensor_dim OOB zero-fill; left/top/bottom are handled by
//     descriptor clipping + LDS pre-zero (edge tiles only).
//   * Separable compute: 5-tap horizontal row sums into LDS, then 5-tap
//     vertical sums + *(1/25), stored with 16B/lane global_store_b128.
// ---------------------------------------------------------------------------

typedef float        f32x4 __attribute__((ext_vector_type(4)));
typedef unsigned int v4u   __attribute__((ext_vector_type(4)));
typedef int          v8i   __attribute__((ext_vector_type(8)));
typedef int          v4i   __attribute__((ext_vector_type(4)));

#define IMG_W 512
#define IMG_H 512
#define TW    64            // output tile width
#define TH    64            // output tile height
#define HW    (TW + 4)      // 68: input tile width incl. halo
#define HH    (TH + 4)      // 68: input tile height incl. halo

__global__ __launch_bounds__(256)
void avefilter5x5_tdm(const float* __restrict__ src, float* __restrict__ dst) {
    __shared__ __align__(16) float sIN [HH * HW];   // 68*68 f32 = 18496 B
    __shared__ __align__(16) float sMID[HH * TW];   // 68*64 f32 = 17408 B

    const int tid = threadIdx.x;
    const int x0  = blockIdx.x * TW;
    const int y0  = blockIdx.y * TH;
    const long img = blockIdx.z;

    const float* simg = src + img * (long)(IMG_W * IMG_H);
    float*       dimg = dst + img * (long)(IMG_W * IMG_H);

    // Pre-zero only when the TDM leaves halo slots unwritten: left edge
    // (x0==0), top edge (y0==0), bottom edge (y0==IMG_H-TH). Right-edge
    // overrun is zero-filled by the TDM itself via tensor_dim0 OOB.
    if (x0 == 0 || y0 == 0 || y0 == IMG_H - TH) {
        f32x4 z = { 0.f, 0.f, 0.f, 0.f };
        for (int i = tid; i < (HH * HW) / 4; i += 256)
            ((f32x4*)sIN)[i] = z;                     // b128 DS stores
    }
    __syncthreads();

    // ---- Wave 0 issues one TDM descriptor for the whole tile -------------
    if (tid < 32) {
#if __HIP_DEVICE_COMPILE__
        const int xs = x0 - 2, ys = y0 - 2;               // halo origin (may be <0)
        const int xlo = xs < 0 ? 0 : xs;
        const int ylo = ys < 0 ? 0 : ys;
        const int lclip = xlo - xs;                       // 0 or 2
        const int tclip = ylo - ys;                       // 0 or 2
        int rows = HH - tclip;
        if (ylo + rows > IMG_H) rows = IMG_H - ylo;       // bottom clip
        const unsigned tile_w = (unsigned)(HW - lclip);   // left clip only
        const unsigned td0 = (unsigned)(IMG_W - xlo);     // OOB-X => zero fill
        const unsigned td1 = (unsigned)(IMG_H - ylo);

        const uint64_t gaddr =
            (uint64_t)(uintptr_t)(simg + (size_t)ylo * IMG_W + xlo);
        const unsigned lds =
            (unsigned)(uintptr_t)(void*)&sIN[tclip * HW + lclip];

        // ---- D# group 0: count=1 | lds_addr | global_addr | type=2 -------
        v4u g0 = { 1u,
                   lds,
                   (unsigned)(gaddr & 0xFFFFFFFFu),
                   (unsigned)((gaddr >> 32) & 0x1FFFFFFu) | (2u << 30) };

        // ---- D# group 1: data_size=4B, iterate_enable, dims/strides ------
        // [17:16]=2 (4B), [19]=1 iterate
        v8i g1 = { (int)((2u << 16) | (1u << 19)),
                   (int)((td0 & 0xFFFFu) << 16),                 // tensor_dim0 lo16 @ [63:48]
                   (int)(((td0 >> 16) & 0xFFFFu)                 // tensor_dim0 hi16
                         | ((td1 & 0xFFFFu) << 16)),             // tensor_dim1 lo16
                   (int)(((td1 >> 16) & 0xFFFFu)                 // tensor_dim1 hi16
                         | ((tile_w & 0xFFFFu) << 16)),          // tile_dim0
                   (int)1,                                       // tile_dim1=1, tile_dim2=0
                   (int)IMG_W,                                   // tensor_dim0_stride lo32
                   0, 0 };

        // ---- D# group 2 (iterate mode): lds/global increments, count -----
        v4i g2 = { 0,
                   (int)HW,                                      // lds_addr_increment (elems)
                   (int)IMG_W,                                   // global_addr_increment lo32
                   (int)(((unsigned)(rows - 1) & 0xFFFFu) << 16) // iterate_count
                 };
        v4i g3 = { 0, 0, 0, 0 };
        v8i g4 = { 0, 0, 0, 0, 0, 0, 0, 0 };                     // unused extra group

        __builtin_amdgcn_tensor_load_to_lds(g0, g1, g2, g3, g4, 0);
        __builtin_amdgcn_s_wait_tensorcnt(0);                    // TENSORcnt -> 0
#endif
    }
    __syncthreads();

    // ---- Horizontal pass: sMID[r][c] = sum sIN[r][c..c+4] ----------------
    // 68 rows x 16 quad-columns = 1088 tasks, 4 outputs each.
    for (int t = tid; t < HH * (TW / 4); t += 256) {
        const int row = t >> 4;
        const int c0  = (t & 15) << 2;
        const float* p = &sIN[row * HW + c0];
        const float a0 = p[0], a1 = p[1], a2 = p[2], a3 = p[3];
        const float a4 = p[4], a5 = p[5], a6 = p[6], a7 = p[7];
        const float s0 = a0 + a1 + a2 + a3 + a4;
        const float s1 = s0 + (a5 - a0);
        const float s2 = s1 + (a6 - a1);
        const float s3 = s2 + (a7 - a2);
        f32x4 r = { s0, s1, s2, s3 };
        *(f32x4*)&sMID[row * TW + c0] = r;
    }
    __syncthreads();

    // ---- Vertical pass + scaled store: 64 rows x 16 quads = 1024 tasks ---
    for (int t = tid; t < TH * (TW / 4); t += 256) {
        const int y  = t >> 4;
        const int c0 = (t & 15) << 2;
        const float* m = &sMID[y * TW + c0];
        f32x4 acc = *(const f32x4*)(m)
                  + *(const f32x4*)(m + 1 * TW)
                  + *(const f32x4*)(m + 2 * TW)
                  + *(const f32x4*)(m + 3 * TW)
                  + *(const f32x4*)(m + 4 * TW);
        acc *= (1.0f / 25.0f);
        *(f32x4*)&dimg[(size_t)(y0 + y) * IMG_W + (x0 + c0)] = acc;   // b128 store
    }
}

extern "C" void kernel_launch(void* const* d_in, const int* in_sizes, int n_in,
                              void* d_out, int out_size, void* d_ws, size_t ws_size,
                              hipStream_t stream) {
    (void)n_in; (void)d_ws; (void)ws_size; (void)out_size;
    const float* x   = (const float*)d_in[0];
    float*       out = (float*)d_out;
    const int images = in_sizes[0] / (IMG_W * IMG_H);   // N*C = 128
    dim3 grid(IMG_W / TW, IMG_H / TH, images);          // 8 x 8 x 128 blocks
    avefilter5x5_tdm<<<grid, 256, 0, stream>>>(x, out);
}